// SAGE_79817672229553
// MI455X (gfx1250) — compile-verified
//
#include <hip/hip_runtime.h>
#include <hip/hip_bf16.h>

typedef float v2f __attribute__((ext_vector_type(2)));
typedef float v8f __attribute__((ext_vector_type(8)));

#define IN_F 128
#define HID_F 128
#define OUT_F 64

// ---------------------------------------------------------------------------
// 1) degree accumulation: deg_out[src[e]] += 1, deg_in[dst[e]] += 1
// ---------------------------------------------------------------------------
__global__ void gcn_degree_kernel(const int* __restrict__ src,
                                  const int* __restrict__ dst,
                                  float* __restrict__ deg_out,
                                  float* __restrict__ deg_in,
                                  int E) {
    int e = blockIdx.x * blockDim.x + threadIdx.x;
    if (e >= E) return;
    atomicAdd(&deg_out[src[e]], 1.0f);
    atomicAdd(&deg_in[dst[e]], 1.0f);
}

// ---------------------------------------------------------------------------
// 2) in-place d[i] = rsqrt(max(d[i], 1))   (runs over both deg arrays at once)
// ---------------------------------------------------------------------------
__global__ void gcn_rsqrt_kernel(float* __restrict__ d, int n) {
    int i = blockIdx.x * blockDim.x + threadIdx.x;
    if (i >= n) return;
    d[i] = rsqrtf(fmaxf(d[i], 1.0f));
}

// ---------------------------------------------------------------------------
// 3) y[node][:] = x[node][:] * rs[node]  (F = 128, float4 per thread)
// ---------------------------------------------------------------------------
__global__ void gcn_scale_kernel(const float* __restrict__ x,
                                 const float* __restrict__ rs,
                                 float* __restrict__ y, int Nn) {
    int tid = blockIdx.x * blockDim.x + threadIdx.x;       // Nn * 32 threads
    int node = tid >> 5;
    int c = tid & 31;
    if (node >= Nn) return;
    float s = rs[node];
    const float4 v = *(const float4*)(x + (size_t)node * 128 + c * 4);
    float4 o = make_float4(v.x * s, v.y * s, v.z * s, v.w * s);
    *(float4*)(y + (size_t)node * 128 + c * 4) = o;
}

// ---------------------------------------------------------------------------
// 4) edge scatter: agg[dst[e]][:] += x[src[e]][:]
//    One float4 chunk per thread; lanes of a wave share an edge (node feature
//    tables are 51 MB -> L2-resident on the 192 MB L2, so these atomics run
//    at L2 rates).
// ---------------------------------------------------------------------------
template <int LOGCHUNKS>
__global__ void gcn_scatter_kernel(const float* __restrict__ x,
                                   float* __restrict__ agg,
                                   const int* __restrict__ src,
                                   const int* __restrict__ dst,
                                   int E) {
    constexpr int F = 4 << LOGCHUNKS;
    long long tid = (long long)blockIdx.x * blockDim.x + threadIdx.x;
    long long e = tid >> LOGCHUNKS;
    int c = (int)(tid & ((1 << LOGCHUNKS) - 1));
    if (e >= E) return;
    int s = src[e];
    int d = dst[e];
    const float4 v = *(const float4*)(x + (size_t)s * F + c * 4);
    float* p = agg + (size_t)d * F + c * 4;
    atomicAdd(p + 0, v.x);
    atomicAdd(p + 1, v.y);
    atomicAdd(p + 2, v.z);
    atomicAdd(p + 3, v.w);
}

// ---------------------------------------------------------------------------
// 5) WMMA GEMM:  Y[Nn x NCOLS] = A[Nn x 128] @ W[128 x NCOLS]
//    EPI==true:  y = relu(y * rs[row] + bias[col])   (layers 1 & 2)
//    EPI==false: plain store                          (layer 3 pre-scatter)
//    V_WMMA_F32_16X16X4_F32 keeps full f32 fidelity; the GEMMs (8.2 GFLOP
//    total) are far from the bottleneck, edge traffic is.
//    Block = 256 threads = 8 waves; each wave owns one 16x16 output tile.
//    W staged in LDS once per block (<= 64 KB).
// ---------------------------------------------------------------------------
template <int NCOLS, bool EPI>
__global__ void gcn_gemm_wmma_kernel(const float* __restrict__ A,
                                     const float* __restrict__ W,
                                     float* __restrict__ Y,
                                     const float* __restrict__ rs,
                                     const float* __restrict__ bias,
                                     int Nn) {
    extern __shared__ float ldsW[];                 // 128 * NCOLS floats
    constexpr int K = 128;

    for (int idx = threadIdx.x; idx < K * NCOLS; idx += blockDim.x)
        ldsW[idx] = W[idx];
    __syncthreads();

    const int wave = threadIdx.x >> 5;
    const int lane = threadIdx.x & 31;
    constexpr int colTiles = NCOLS >> 4;            // 8 (NCOLS=128) or 4 (64)
    constexpr int rowsPerBlock = (8 / colTiles) * 16;  // 16 or 32
    const int ct = wave % colTiles;
    const int rt = wave / colTiles;
    const int row0 = blockIdx.x * rowsPerBlock + rt * 16;
    const int col0 = ct * 16;
    if (row0 >= Nn) return;                         // wave-uniform: EXEC stays full

    const int m = lane & 15;      // A row within tile / B,C column within tile
    const int half = lane >> 4;

    const float* arow = A + (size_t)(row0 + m) * K;

    v8f c = {0.f, 0.f, 0.f, 0.f, 0.f, 0.f, 0.f, 0.f};

#pragma unroll 4
    for (int k0 = 0; k0 < K; k0 += 4) {
        const int kk = k0 + 2 * half;
        // A 16x4 f32 fragment: VGPR0 = A[m][kk], VGPR1 = A[m][kk+1]
        const float2 av = *(const float2*)(arow + kk);
        v2f a;
        a.x = av.x;
        a.y = av.y;
        // B 4x16 f32 fragment (mirrored layout): VGPRv = W[kk+v][col0+m]
        v2f b;
        b.x = ldsW[kk * NCOLS + col0 + m];
        b.y = ldsW[(kk + 1) * NCOLS + col0 + m];
        c = __builtin_amdgcn_wmma_f32_16x16x4_f32(
            /*neg_a=*/false, a, /*neg_b=*/false, b,
            /*c_mod=*/(short)0, c, /*reuse_a=*/false, /*reuse_b=*/false);
    }

    // C/D layout: lane l holds column col0 + (l&15); VGPR i holds row i + 8*(l>>4)
    float* yout = Y + (size_t)(row0 + 8 * half) * NCOLS + col0 + m;
    if (EPI) {
        const float bcol = bias[col0 + m];                       // 1 load, invariant
        const float4 r0 = *(const float4*)(rs + row0 + 8 * half);      // 8 contiguous
        const float4 r1 = *(const float4*)(rs + row0 + 8 * half + 4);  // rs values
        const float rsv[8] = {r0.x, r0.y, r0.z, r0.w, r1.x, r1.y, r1.z, r1.w};
#pragma unroll
        for (int i = 0; i < 8; ++i) {
            float v = fmaf(c[i], rsv[i], bcol);
            v = fmaxf(v, 0.f);
            yout[(size_t)i * NCOLS] = v;
        }
    } else {
#pragma unroll
        for (int i = 0; i < 8; ++i)
            yout[(size_t)i * NCOLS] = c[i];
    }
}

// ---------------------------------------------------------------------------
// 6) final epilogue (layer 3, OUT=64): out = agg * rs_in[node] + b3
// ---------------------------------------------------------------------------
__global__ void gcn_final_kernel(const float* __restrict__ agg,
                                 const float* __restrict__ rs_in,
                                 const float* __restrict__ b3,
                                 float* __restrict__ out, int Nn) {
    int tid = blockIdx.x * blockDim.x + threadIdx.x;        // Nn * 16 threads
    int node = tid >> 4;
    int c = tid & 15;
    if (node >= Nn) return;
    float s = rs_in[node];
    const float4 v = *(const float4*)(agg + (size_t)node * 64 + c * 4);
    const float4 bb = *(const float4*)(b3 + c * 4);
    float4 o = make_float4(fmaf(v.x, s, bb.x), fmaf(v.y, s, bb.y),
                           fmaf(v.z, s, bb.z), fmaf(v.w, s, bb.w));
    *(float4*)(out + (size_t)node * 64 + c * 4) = o;
}

// ---------------------------------------------------------------------------
extern "C" void kernel_launch(void* const* d_in, const int* in_sizes, int n_in,
                              void* d_out, int out_size, void* d_ws, size_t ws_size,
                              hipStream_t stream) {
    const float* h  = (const float*)d_in[0];
    const int*   src = (const int*)d_in[1];
    const int*   dst = (const int*)d_in[2];
    const float* W1 = (const float*)d_in[3];
    const float* b1 = (const float*)d_in[4];
    const float* W2 = (const float*)d_in[5];
    const float* b2 = (const float*)d_in[6];
    const float* W3 = (const float*)d_in[7];
    const float* b3 = (const float*)d_in[8];
    float* out = (float*)d_out;

    const int Nn = in_sizes[0] / IN_F;   // 100000
    const int E  = in_sizes[1];          // 1600000

    // workspace layout (floats): rs_out[N] | rs_in[N] | bufA[N*128] | bufB[N*128]
    float* rs_out = (float*)d_ws;
    float* rs_in  = rs_out + Nn;
    float* bufA   = rs_in + Nn;
    float* bufB   = bufA + (size_t)Nn * 128;

    const int TB = 256;
    auto cdiv = [](long long a, long long b) { return (unsigned)((a + b - 1) / b); };

    // --- degrees ---
    hipMemsetAsync(rs_out, 0, (size_t)2 * Nn * sizeof(float), stream);
    gcn_degree_kernel<<<cdiv(E, TB), TB, 0, stream>>>(src, dst, rs_out, rs_in, E);
    gcn_rsqrt_kernel<<<cdiv(2LL * Nn, TB), TB, 0, stream>>>(rs_out, 2 * Nn);

    const unsigned scaleGrid   = cdiv((long long)Nn * 32, TB);
    const unsigned scat128Grid = cdiv((long long)E * 32, TB);
    const unsigned scat64Grid  = cdiv((long long)E * 16, TB);
    const unsigned gemm128Grid = cdiv(Nn, 16);
    const unsigned gemm64Grid  = cdiv(Nn, 32);
    const size_t   lds128 = (size_t)128 * 128 * sizeof(float);  // 64 KB
    const size_t   lds64  = (size_t)128 * 64 * sizeof(float);   // 32 KB

    // --- layer 1: scale -> scatter -> gemm(+scale,bias,relu) ---
    gcn_scale_kernel<<<scaleGrid, TB, 0, stream>>>(h, rs_out, bufA, Nn);
    hipMemsetAsync(bufB, 0, (size_t)Nn * 128 * sizeof(float), stream);
    gcn_scatter_kernel<5><<<scat128Grid, TB, 0, stream>>>(bufA, bufB, src, dst, E);
    gcn_gemm_wmma_kernel<128, true><<<gemm128Grid, TB, lds128, stream>>>(
        bufB, W1, bufA, rs_in, b1, Nn);

    // --- layer 2 ---
    gcn_scale_kernel<<<scaleGrid, TB, 0, stream>>>(bufA, rs_out, bufB, Nn);
    hipMemsetAsync(bufA, 0, (size_t)Nn * 128 * sizeof(float), stream);
    gcn_scatter_kernel<5><<<scat128Grid, TB, 0, stream>>>(bufB, bufA, src, dst, E);
    gcn_gemm_wmma_kernel<128, true><<<gemm128Grid, TB, lds128, stream>>>(
        bufA, W2, bufB, rs_in, b2, Nn);

    // --- layer 3: scale -> gemm (128->64, no epilogue) -> scatter -> epilogue ---
    gcn_scale_kernel<<<scaleGrid, TB, 0, stream>>>(bufB, rs_out, bufA, Nn);
    gcn_gemm_wmma_kernel<64, false><<<gemm64Grid, TB, lds64, stream>>>(
        bufA, W3, bufB, nullptr, nullptr, Nn);
    hipMemsetAsync(bufA, 0, (size_t)Nn * 64 * sizeof(float), stream);
    gcn_scatter_kernel<4><<<scat64Grid, TB, 0, stream>>>(bufB, bufA, src, dst, E);
    gcn_final_kernel<<<cdiv((long long)Nn * 16, TB), TB, 0, stream>>>(
        bufA, rs_in, b3, out, Nn);
}